// LGBayesianNetwork_79680233275469
// MI455X (gfx1250) — compile-verified
//
#include <hip/hip_runtime.h>

typedef __attribute__((ext_vector_type(2))) float v2f;
typedef __attribute__((ext_vector_type(8))) float v8f;

#define S_ 8
#define F_ 128
#define L_ 65536
#define MAXAR_ 3
#define T_ (L_ - MAXAR_)      // 65533
#define WROW_ (F_ + MAXAR_)   // 131

// ---------------------------------------------------------------------------
// Prep: build (a) masked parent weights, transposed + swizzled for WMMA-B
// loads:  swz[((s*32 + fc)*128 + k)*4 + j] = W[s,k,4*fc+j] * G[s,k,4*fc+j]
// and (b) per-(s,k) {arw1, arw2, arw3, bias} with AR-order mask applied.
// ---------------------------------------------------------------------------
__global__ __launch_bounds__(256) void prep_kernel(
    const float* __restrict__ w, const int* __restrict__ g,
    const int* __restrict__ aror, float* __restrict__ swz,
    float* __restrict__ arwb) {
  int e = blockIdx.x * 256 + threadIdx.x;   // 0 .. 131071
  int j  = e & 3;
  int k  = (e >> 2) & 127;
  int fc = (e >> 9) & 31;
  int s  = e >> 14;
  int f  = 4 * fc + j;
  float wv = w[(s * F_ + k) * WROW_ + f];
  int   gv = g[(s * F_ + k) * F_ + f];
  swz[e] = gv ? wv : 0.0f;

  if (e < S_ * F_) {                        // 1024 threads build AR/bias table
    int s2 = e >> 7, k2 = e & 127;
    int base = (s2 * F_ + k2) * WROW_;
    int ar = aror[s2 * F_ + k2];
    arwb[e * 4 + 0] = (ar >= 1) ? w[base + F_ + 0] : 0.0f;  // lag j=1 -> x[t+2]
    arwb[e * 4 + 1] = (ar >= 2) ? w[base + F_ + 1] : 0.0f;  // lag j=2 -> x[t+1]
    arwb[e * 4 + 2] = (ar >= 3) ? w[base + F_ + 2] : 0.0f;  // lag j=3 -> x[t]
    arwb[e * 4 + 3] = w[base + k2];                         // bias = diagonal
  }
}

// ---------------------------------------------------------------------------
// Main kernel: grid (512, S). 256 threads = 8 waves; wave computes a
// 16(T) x 128(K) output tile for graph s = blockIdx.y via 32x8 chained
// v_wmma_f32_16x16x4_f32, then fuses bias + AR terms in the epilogue.
// ---------------------------------------------------------------------------
__global__ __launch_bounds__(256) void bn_gemm_kernel(
    const float* __restrict__ x, const float* __restrict__ swz,
    const float* __restrict__ arwb, float* __restrict__ out) {
  __shared__ float lW[32 * 128 * 4];        // 64 KB: [fc][k][j]

  const int s = blockIdx.y;

  // Stage this graph's swizzled weights into LDS (coalesced 16B copies).
  {
    const float4* src = (const float4*)(swz + (size_t)s * (F_ * F_));
    float4* dst = (float4*)lW;
#pragma unroll
    for (int i = 0; i < 16; ++i)
      dst[threadIdx.x + i * 256] = src[threadIdx.x + i * 256];
  }
  __syncthreads();

  const int lane = threadIdx.x & 31;
  const int wave = threadIdx.x >> 5;
  const int m    = lane & 15;               // row within tile / col within n-tile
  const int half = lane >> 4;               // lane-half: K offset 2, M offset 8

  const int t0 = (blockIdx.x * 8 + wave) * 16;

  // A operand: lane holds row t0+m of xs (= x row t+3), K-pair (2*half, 2*half+1)
  int tA = t0 + m;
  if (tA > T_ - 1) tA = T_ - 1;             // clamp (WMMA needs EXEC all-1s)
  const float* xp = x + (size_t)(tA + MAXAR_) * F_ + 2 * half;

  v8f acc[8];
#pragma unroll
  for (int n = 0; n < 8; ++n) acc[n] = (v8f)0.0f;

  for (int fc = 0; fc < 32; ++fc) {
    v2f a = *(const v2f*)(xp + 4 * fc);     // 8B global load, 8B aligned
    const float* wb = lW + (fc * 128 + m) * 4 + 2 * half;
#pragma unroll
    for (int n = 0; n < 8; ++n) {
      v2f b = *(const v2f*)(wb + n * 64);   // ds_load_b64
      acc[n] = __builtin_amdgcn_wmma_f32_16x16x4_f32(
          false, a, false, b, (short)0, acc[n], false, false);
    }
  }

  // Epilogue: D VGPR r -> row t0 + r + 8*half, col k = n*16 + m.
  const int tbase = t0 + half * 8;
#pragma unroll
  for (int n = 0; n < 8; ++n) {
    const int k = n * 16 + m;
    const float4 aw = ((const float4*)arwb)[s * F_ + k];  // {arw1,arw2,arw3,bias}

    float xl[10];                           // x[tbase .. tbase+9][k] (lag column)
#pragma unroll
    for (int r = 0; r < 10; ++r) {
      int tt = tbase + r;
      if (tt > L_ - 1) tt = L_ - 1;         // clamped; unused lanes never stored
      xl[r] = x[(size_t)tt * F_ + k];
    }
#pragma unroll
    for (int r = 0; r < 8; ++r) {
      int t = tbase + r;
      if (t < T_) {
        float v = acc[n][r] + aw.w + aw.x * xl[r + 2] + aw.y * xl[r + 1]
                + aw.z * xl[r];
        // Streaming 268MB result: non-temporal so x/W stay resident in L2.
        __builtin_nontemporal_store(v, out + ((size_t)s * T_ + t) * F_ + k);
      }
    }
  }
}

// ---------------------------------------------------------------------------
extern "C" void kernel_launch(void* const* d_in, const int* in_sizes, int n_in,
                              void* d_out, int out_size, void* d_ws, size_t ws_size,
                              hipStream_t stream) {
  (void)in_sizes; (void)n_in; (void)out_size; (void)ws_size;
  const float* x    = (const float*)d_in[0];
  const float* w    = (const float*)d_in[1];
  const int*   g    = (const int*)d_in[2];
  const int*   aror = (const int*)d_in[3];
  // d_in[4] = maxar (compile-time constant 3 here)

  float* swz  = (float*)d_ws;               // 131072 floats = 512 KB
  float* arwb = swz + S_ * F_ * F_;         // 4096 floats   = 16 KB

  prep_kernel<<<dim3(512), dim3(256), 0, stream>>>(w, g, aror, swz, arwb);
  bn_gemm_kernel<<<dim3(512, S_), dim3(256), 0, stream>>>(x, swz, arwb,
                                                          (float*)d_out);
}